// GATModel_32925219291644
// MI455X (gfx1250) — compile-verified
//
#include <hip/hip_runtime.h>
#include <math.h>

typedef float v2f __attribute__((ext_vector_type(2)));
typedef float v8f __attribute__((ext_vector_type(8)));

#define LEAKY 0.2f

// ---------------------------------------------------------------- utilities
__global__ void fill_f32(float* __restrict__ p, long n, float v) {
  long i = (long)blockIdx.x * blockDim.x + threadIdx.x;
  if (i < n) p[i] = v;
}

__device__ __forceinline__ void atomicMaxF(float* addr, float val) {
  // monotone int mapping trick; slots init'ed to -inf (0xFF800000)
  if (val >= 0.0f) atomicMax((int*)addr, __float_as_int(val));
  else             atomicMin((unsigned int*)addr, (unsigned int)__float_as_int(val));
}

// ---------------------------------------------------------------- WMMA GEMM
// C[M, NT*16] = A[M, K] * B[K, NT*16], fp32 via V_WMMA_F32_16X16X4_F32.
// One wave -> 16 output rows, NT accumulators of 16 cols each.
template <int K, int NT>
__global__ __launch_bounds__(256) void gemm_wmma(const float* __restrict__ A,
                                                 const float* __restrict__ B,
                                                 float* __restrict__ C, int M) {
  const int wave = threadIdx.x >> 5;
  const int lane = threadIdx.x & 31;
  const int tile = blockIdx.x * 8 + wave;          // 8 waves / block
  if (tile * 16 >= M) return;                      // wave-uniform: EXEC stays all-1
  const int row0 = tile * 16;
  const int half = lane >> 4;                      // 0: lanes 0-15, 1: lanes 16-31
  const int l16  = lane & 15;
  const int NCOL = NT * 16;
  const int arow = row0 + l16;

  v8f zero = {};
  v8f acc[NT];
#pragma unroll
  for (int t = 0; t < NT; ++t) acc[t] = zero;

  for (int k = 0; k < K; k += 4) {
    // A fragment (16x4): VGPR0 = K=k (lanes0-15) / k+2 (lanes16-31); VGPR1 = k+1 / k+3
    v2f a;
    a.x = A[(long)arow * K + k + 2 * half + 0];
    a.y = A[(long)arow * K + k + 2 * half + 1];
#pragma unroll
    for (int t = 0; t < NT; ++t) {
      const int col = t * 16 + l16;
      // B fragment (4x16), symmetric layout: rows striped across lanes per VGPR
      v2f b;
      b.x = B[(k + 2 * half + 0) * NCOL + col];
      b.y = B[(k + 2 * half + 1) * NCOL + col];
      acc[t] = __builtin_amdgcn_wmma_f32_16x16x4_f32(
          false, a, false, b, (short)0, acc[t], false, false);
    }
  }

  // C/D layout: VGPR r -> M=row0+r (lanes 0-15) / M=row0+r+8 (lanes 16-31), N=l16
#pragma unroll
  for (int t = 0; t < NT; ++t) {
    const int col = t * 16 + l16;
#pragma unroll
    for (int r = 0; r < 8; ++r) {
      const int row = row0 + half * 8 + r;
      C[(long)row * NCOL + col] = acc[t][r];
    }
  }
}

// ---------------------------------------------------------------- attention
// alpha_src[n,h] = sum_d z[n,h*Dh+d]*a_src[h,d] ; same for dst. One thread/(n,h).
__global__ void alpha_kernel(const float* __restrict__ z,
                             const float* __restrict__ a_src,
                             const float* __restrict__ a_dst,
                             float* __restrict__ as, float* __restrict__ ad,
                             int N, int H, int Dh) {
  int t = blockIdx.x * blockDim.x + threadIdx.x;
  if (t >= N * H) return;
  int n = t / H, h = t % H;
  const float* zp = z + (long)n * H * Dh + h * Dh;
  const float* pa = a_src + h * Dh;
  const float* pb = a_dst + h * Dh;
  float s = 0.f, d = 0.f;
  for (int i = 0; i < Dh; ++i) { float v = zp[i]; s += v * pa[i]; d += v * pb[i]; }
  as[t] = s;
  ad[t] = d;
}

__device__ __forceinline__ float edge_e(const float* as, const float* ad,
                                        int s, int d, int h, int H) {
  float v = as[s * H + h] + ad[d * H + h];
  return v >= 0.f ? v : LEAKY * v;
}

// pass 1: per-(dst,head) running max
__global__ void edge_max(const int* __restrict__ src, const int* __restrict__ dst,
                         const float* __restrict__ as, const float* __restrict__ ad,
                         float* __restrict__ m, long EH, int H) {
  long t = (long)blockIdx.x * blockDim.x + threadIdx.x;
  if (t >= EH) return;
  int e = (int)(t / H), h = (int)(t % H);
  int s = src[e], d = dst[e];
  atomicMaxF(&m[(long)d * H + h], edge_e(as, ad, s, d, h, H));
}

// pass 2: denom[dst,h] += exp(e - m[dst,h])
__global__ void edge_denom(const int* __restrict__ src, const int* __restrict__ dst,
                           const float* __restrict__ as, const float* __restrict__ ad,
                           const float* __restrict__ m, float* __restrict__ den,
                           long EH, int H) {
  long t = (long)blockIdx.x * blockDim.x + threadIdx.x;
  if (t >= EH) return;
  int e = (int)(t / H), h = (int)(t % H);
  int s = src[e], d = dst[e];
  float v = edge_e(as, ad, s, d, h, H);
  atomicAdd(&den[(long)d * H + h], expf(v - m[(long)d * H + h]));
}

// pass 3: agg[dst,c] += z[src,c] * alpha(e,h).  One thread per (edge, channel).
__global__ void edge_agg(const int* __restrict__ src, const int* __restrict__ dst,
                         const float* __restrict__ as, const float* __restrict__ ad,
                         const float* __restrict__ m, const float* __restrict__ den,
                         const float* __restrict__ z, float* __restrict__ agg,
                         long ED, int D, int Dh, int H) {
  long t = (long)blockIdx.x * blockDim.x + threadIdx.x;
  if (t >= ED) return;
  int e = (int)(t / D), c = (int)(t % D);
  int h = c / Dh;
  int s = src[e], d = dst[e];
  float v = edge_e(as, ad, s, d, h, H);
  float w = expf(v - m[(long)d * H + h]);
  float alpha = w / (den[(long)d * H + h] + 1e-16f);
  atomicAdd(&agg[(long)d * D + c], z[(long)s * D + c] * alpha);
}

// ---------------------------------------------------------------- epilogues
__global__ void bias_elu(float* __restrict__ p, const float* __restrict__ b,
                         long n, int D) {
  long i = (long)blockIdx.x * blockDim.x + threadIdx.x;
  if (i >= n) return;
  float v = p[i] + b[i % D];
  p[i] = v > 0.f ? v : expm1f(v);
}

__global__ void bias_add(float* __restrict__ p, const float* __restrict__ b,
                         long n, int D) {
  long i = (long)blockIdx.x * blockDim.x + threadIdx.x;
  if (i >= n) return;
  p[i] += b[i % D];
}

// ---------------------------------------------------------------- launcher
extern "C" void kernel_launch(void* const* d_in, const int* in_sizes, int n_in,
                              void* d_out, int out_size, void* d_ws, size_t ws_size,
                              hipStream_t stream) {
  const float* x   = (const float*)d_in[0];
  const int*   ei  = (const int*)d_in[1];
  const float* W1  = (const float*)d_in[2];
  const float* aS1 = (const float*)d_in[3];
  const float* aD1 = (const float*)d_in[4];
  const float* b1  = (const float*)d_in[5];
  const float* W2  = (const float*)d_in[6];
  const float* aS2 = (const float*)d_in[7];
  const float* aD2 = (const float*)d_in[8];
  const float* b2  = (const float*)d_in[9];
  float* out = (float*)d_out;

  const int N = in_sizes[0] / 128;       // 50000 nodes
  const int E = in_sizes[1] / 2;         // 850000 edges
  const int* src = ei;                   // edge_index[0]
  const int* dst = ei + E;               // edge_index[1]

  // workspace carve (floats)
  float* w = (float*)d_ws;
  float* z1   = w; w += (long)N * 64;
  float* as1  = w; w += (long)N * 8;
  float* ad1  = w; w += (long)N * 8;
  float* m1   = w; w += (long)N * 8;
  float* den1 = w; w += (long)N * 8;
  float* agg1 = w; w += (long)N * 64;    // becomes h after bias+ELU (in-place)
  float* z2   = w; w += (long)N * 16;
  float* as2  = w; w += (long)N;
  float* ad2  = w; w += (long)N;
  float* m2   = w; w += (long)N;
  float* den2 = w; w += (long)N;

  auto nb = [](long n) { return (unsigned)((n + 255) / 256); };

  // ---- per-call init (scratch is poisoned; must be deterministic each call)
  fill_f32<<<nb((long)N * 8),  256, 0, stream>>>(m1,  (long)N * 8,  -INFINITY);
  fill_f32<<<nb((long)N * 8),  256, 0, stream>>>(den1,(long)N * 8,  0.f);
  fill_f32<<<nb((long)N * 64), 256, 0, stream>>>(agg1,(long)N * 64, 0.f);
  fill_f32<<<nb(N),            256, 0, stream>>>(m2,  N,            -INFINITY);
  fill_f32<<<nb(N),            256, 0, stream>>>(den2,N,            0.f);
  fill_f32<<<nb((long)N * 16), 256, 0, stream>>>(out, (long)N * 16, 0.f);

  // ---- layer 1: z1 = x @ W1 (WMMA f32), heads H=8, Dh=8
  {
    const int tiles = (N + 15) / 16;                 // 3125 (divides exactly)
    gemm_wmma<128, 4><<<(tiles + 7) / 8, 256, 0, stream>>>(x, W1, z1, N);
    alpha_kernel<<<nb((long)N * 8), 256, 0, stream>>>(z1, aS1, aD1, as1, ad1, N, 8, 8);
    edge_max  <<<nb((long)E * 8),  256, 0, stream>>>(src, dst, as1, ad1, m1, (long)E * 8, 8);
    edge_denom<<<nb((long)E * 8),  256, 0, stream>>>(src, dst, as1, ad1, m1, den1, (long)E * 8, 8);
    edge_agg  <<<nb((long)E * 64), 256, 0, stream>>>(src, dst, as1, ad1, m1, den1,
                                                     z1, agg1, (long)E * 64, 64, 8, 8);
    bias_elu  <<<nb((long)N * 64), 256, 0, stream>>>(agg1, b1, (long)N * 64, 64);
  }

  // ---- layer 2: z2 = h @ W2 (WMMA f32), single head, Dh=16
  {
    const int tiles = (N + 15) / 16;
    gemm_wmma<64, 1><<<(tiles + 7) / 8, 256, 0, stream>>>(agg1, W2, z2, N);
    alpha_kernel<<<nb(N), 256, 0, stream>>>(z2, aS2, aD2, as2, ad2, N, 1, 16);
    edge_max  <<<nb(E),            256, 0, stream>>>(src, dst, as2, ad2, m2, E, 1);
    edge_denom<<<nb(E),            256, 0, stream>>>(src, dst, as2, ad2, m2, den2, E, 1);
    edge_agg  <<<nb((long)E * 16), 256, 0, stream>>>(src, dst, as2, ad2, m2, den2,
                                                     z2, out, (long)E * 16, 16, 16, 1);
    bias_add  <<<nb((long)N * 16), 256, 0, stream>>>(out, b2, (long)N * 16, 16);
  }
}